// SynMarginLoss_11897059410047
// MI455X (gfx1250) — compile-verified
//
#include <hip/hip_runtime.h>
#include <math.h>

typedef __attribute__((ext_vector_type(2))) float v2f;
typedef __attribute__((ext_vector_type(8))) float v8f;

#define B_ROWS      1024
#define N_COLS      100000
#define D_DIM       64
#define CHUNK       4096
#define NCHUNK      25          // ceil(100000 / 4096)
#define KB          64          // n-values staged per LDS block
#define ROWS_PER_WG 32
#define WST         68          // weight-tile LDS row stride (floats) -> conflict-free A reads
#define EST         72          // E-tile LDS row stride (floats)      -> conflict-free B reads
#define EPSV        1e-8f
#define MARGINV     0.1f

// ---------------- Kernel 1: per-row max of pred ----------------
__global__ __launch_bounds__(256) void rowmax_kernel(const float* __restrict__ pred,
                                                     float* __restrict__ rowmax) {
    int b = blockIdx.x;
    const float* row = pred + (size_t)b * N_COLS;
    float m = -INFINITY;
    for (int i = threadIdx.x; i < N_COLS; i += 256) m = fmaxf(m, row[i]);
    __shared__ float sm[256];
    sm[threadIdx.x] = m;
    __syncthreads();
    for (int s = 128; s > 0; s >>= 1) {
        if (threadIdx.x < s) sm[threadIdx.x] = fmaxf(sm[threadIdx.x], sm[threadIdx.x + s]);
        __syncthreads();
    }
    if (threadIdx.x == 0) rowmax[b] = sm[0];
}

// ---------------- Kernel 2: fused weighted GEMMs via f32 WMMA ----------------
// grid = (B_ROWS/ROWS_PER_WG, NCHUNK), block = 256 (8 waves).
// Wave w: row half-tile rh = w>>2 (16 rows), d-slice dw = w&3 (16 cols).
__global__ __launch_bounds__(256) void gemm_kernel(const float* __restrict__ pred,
                                                   const float* __restrict__ target,
                                                   const float* __restrict__ ev,
                                                   const float* __restrict__ rowmax,
                                                   float* __restrict__ ppred,
                                                   float* __restrict__ ptgt) {
    __shared__ float wp[ROWS_PER_WG * WST];   // exp(pred - rowmax) tile
    __shared__ float wt[ROWS_PER_WG * WST];   // relu(target) tile
    __shared__ float et[KB * EST];            // E tile [n][d]
    __shared__ float smax[ROWS_PER_WG];

    const int r0 = blockIdx.x * ROWS_PER_WG;
    const int ch = blockIdx.y;
    const int t  = threadIdx.x;

    if (t < ROWS_PER_WG) smax[t] = rowmax[r0 + t];

    const int lane  = t & 31;
    const int wave  = t >> 5;
    const int rh    = wave >> 2;              // 0..1
    const int dw    = wave & 3;               // 0..3
    const int mrow  = lane & 15;              // A-matrix M (within 16)
    const int khalf = (lane >> 4) << 1;       // 0 or 2 (K offset per ISA layout)
    const int dcol  = dw * 16 + (lane & 15);  // B/C column

    v8f cp = {0.f,0.f,0.f,0.f,0.f,0.f,0.f,0.f};
    v8f ct = {0.f,0.f,0.f,0.f,0.f,0.f,0.f,0.f};

    for (int nb = 0; nb < CHUNK / KB; ++nb) {
        const int nbase = ch * CHUNK + nb * KB;
        __syncthreads();   // previous compute done (also publishes smax at nb==0)

        const bool full = (nbase + KB) <= N_COLS;
        // Stage weight tiles: 32 rows x 64 cols, 512 float4 slots over 256 threads
        #pragma unroll
        for (int j = 0; j < 2; ++j) {
            const int q   = t + j * 256;
            const int row = q >> 4;
            const int c4  = (q & 15) * 4;
            const size_t g = (size_t)(r0 + row) * N_COLS + nbase + c4;
            const float  mx = smax[row];
            if (full) {
                const float4 pv = *(const float4*)(pred + g);
                const float4 tv = *(const float4*)(target + g);
                float4 wpv = { __expf(pv.x - mx), __expf(pv.y - mx),
                               __expf(pv.z - mx), __expf(pv.w - mx) };
                float4 wtv = { fmaxf(tv.x, 0.f), fmaxf(tv.y, 0.f),
                               fmaxf(tv.z, 0.f), fmaxf(tv.w, 0.f) };
                *(float4*)(wp + row * WST + c4) = wpv;
                *(float4*)(wt + row * WST + c4) = wtv;
            } else {
                #pragma unroll
                for (int e = 0; e < 4; ++e) {
                    const int n = nbase + c4 + e;
                    float pw = 0.f, twv = 0.f;
                    if (n < N_COLS) {
                        pw  = __expf(pred[g + e] - mx);
                        twv = fmaxf(target[g + e], 0.f);
                    }
                    wp[row * WST + c4 + e] = pw;
                    wt[row * WST + c4 + e] = twv;
                }
            }
        }
        // Stage E tile: 64 rows x 64 cols, 1024 float4 slots over 256 threads
        #pragma unroll
        for (int j = 0; j < 4; ++j) {
            const int q   = t + j * 256;
            const int row = q >> 4;
            const int c4  = (q & 15) * 4;
            const int n   = nbase + row;
            float4 v = {0.f, 0.f, 0.f, 0.f};
            if (n < N_COLS) v = *(const float4*)(ev + (size_t)n * D_DIM + c4);
            *(float4*)(et + row * EST + c4) = v;
        }
        __syncthreads();

        // 16 K-steps of 4: two V_WMMA_F32_16X16X4_F32 per step (pred & tgt share B)
        #pragma unroll
        for (int kk = 0; kk < KB / 4; ++kk) {
            const int kb   = kk * 4 + khalf;
            const int aoff = (rh * 16 + mrow) * WST + kb;   // even -> 8B aligned
            v2f ap = *(const v2f*)(wp + aoff);
            v2f at = *(const v2f*)(wt + aoff);
            v2f bv;
            bv.x = et[(kb    ) * EST + dcol];
            bv.y = et[(kb + 1) * EST + dcol];
            cp = __builtin_amdgcn_wmma_f32_16x16x4_f32(false, ap, false, bv, (short)0, cp, false, false);
            ct = __builtin_amdgcn_wmma_f32_16x16x4_f32(false, at, false, bv, (short)0, ct, false, false);
        }
    }

    // C/D layout: VGPR i -> M=i (lanes 0-15) / M=i+8 (lanes 16-31), N = lane&15
    const int brow_base = r0 + rh * 16 + ((lane >> 4) << 3);
    #pragma unroll
    for (int i = 0; i < 8; ++i) {
        const size_t o = ((size_t)ch * B_ROWS + (brow_base + i)) * D_DIM + dcol;
        ppred[o] = cp[i];
        ptgt[o]  = ct[i];
    }
}

// ---------------- Kernel 3: per-row finalize (1 wave per row) ----------------
__device__ __forceinline__ float wave_sum(float v) {
    #pragma unroll
    for (int off = 16; off > 0; off >>= 1) v += __shfl_xor(v, off, 32);
    return v;
}

__global__ __launch_bounds__(32) void finalize_rows(const float* __restrict__ ppred,
                                                    const float* __restrict__ ptgt,
                                                    float* __restrict__ rowloss) {
    const int b = blockIdx.x, lane = threadIdx.x;
    float sp0 = 0.f, sp1 = 0.f, st0 = 0.f, st1 = 0.f;
    for (int c = 0; c < NCHUNK; ++c) {
        const size_t o = ((size_t)c * B_ROWS + b) * D_DIM;
        sp0 += ppred[o + lane];      sp1 += ppred[o + lane + 32];
        st0 += ptgt[o + lane];       st1 += ptgt[o + lane + 32];
    }
    const float np  = wave_sum(sp0 * sp0 + sp1 * sp1);
    const float ip  = 1.f / fmaxf(sqrtf(np), EPSV);
    const float pe0 = sp0 * ip, pe1 = sp1 * ip;
    const float nt  = wave_sum(st0 * st0 + st1 * st1);
    const float it  = 1.f / fmaxf(sqrtf(nt), EPSV);
    const float te0 = st0 * it, te1 = st1 * it;

    const float dot = wave_sum(pe0 * te0 + pe1 * te1);   // pos_sim
    float ng0 = pe0 - dot * te0, ng1 = pe1 - dot * te1;
    float nn  = wave_sum(ng0 * ng0 + ng1 * ng1);
    if (sqrtf(nn) <= EPSV) {                              // degenerate projection
        ng0 = pe0 - te0; ng1 = pe1 - te1;
        nn  = wave_sum(ng0 * ng0 + ng1 * ng1);
    }
    const float in2 = 1.f / fmaxf(sqrtf(nn), EPSV);
    ng0 *= in2; ng1 *= in2;
    const float negsim = wave_sum(pe0 * ng0 + pe1 * ng1);
    if (lane == 0) rowloss[b] = fmaxf(0.f, MARGINV + negsim - dot);
}

// ---------------- Kernel 4: deterministic mean ----------------
__global__ __launch_bounds__(256) void reduce_mean(const float* __restrict__ rowloss,
                                                   float* __restrict__ out) {
    __shared__ float s[256];
    float v = 0.f;
    for (int i = threadIdx.x; i < B_ROWS; i += 256) v += rowloss[i];
    s[threadIdx.x] = v;
    __syncthreads();
    for (int k = 128; k > 0; k >>= 1) {
        if (threadIdx.x < k) s[threadIdx.x] += s[threadIdx.x + k];
        __syncthreads();
    }
    if (threadIdx.x == 0) out[0] = s[0] * (1.0f / B_ROWS);
}

extern "C" void kernel_launch(void* const* d_in, const int* in_sizes, int n_in,
                              void* d_out, int out_size, void* d_ws, size_t ws_size,
                              hipStream_t stream) {
    const float* pred   = (const float*)d_in[0];   // [1024, 100000]
    const float* target = (const float*)d_in[1];   // [1024, 100000]
    const float* ev     = (const float*)d_in[2];   // [100000, 64]

    float* ws      = (float*)d_ws;
    float* rowmax  = ws;                                   // 1024
    float* rowloss = ws + 1024;                            // 1024
    float* ppred   = ws + 2048;                            // NCHUNK*1024*64
    float* ptgt    = ppred + (size_t)NCHUNK * B_ROWS * D_DIM;

    rowmax_kernel<<<B_ROWS, 256, 0, stream>>>(pred, rowmax);
    gemm_kernel<<<dim3(B_ROWS / ROWS_PER_WG, NCHUNK), 256, 0, stream>>>(
        pred, target, ev, rowmax, ppred, ptgt);
    finalize_rows<<<B_ROWS, 32, 0, stream>>>(ppred, ptgt, rowloss);
    reduce_mean<<<1, 256, 0, stream>>>(rowloss, (float*)d_out);
}